// SMPLXHeadCamFull_32177894981692
// MI455X (gfx1250) — compile-verified
//
#include <hip/hip_runtime.h>
#include <hip/hip_bf16.h>
#include <math.h>

// ---------------- problem constants ----------------
#define BATCH 1024
#define NJ    127
#define NV    10475
#define FEAT  479          // 52*9 + 11
#define KPAD  480          // FEAT padded to a multiple of 4 (zero column)
#define NVCOL (NV*3)       // 31425
#define NJCOL (NJ*3)       // 381

typedef __attribute__((ext_vector_type(2))) float v2f;
typedef __attribute__((ext_vector_type(8))) float v8f;

// ---------------- feat assembly ----------------
__global__ void build_feat_kernel(const float* __restrict__ body,
                                  const float* __restrict__ lh,
                                  const float* __restrict__ rh,
                                  const float* __restrict__ shp,
                                  float* __restrict__ feat) {
  int idx = blockIdx.x * blockDim.x + threadIdx.x;
  if (idx >= BATCH * KPAD) return;
  int b = idx / KPAD;
  int c = idx - b * KPAD;
  float v;
  if (c < 198)      v = body[b * 198 + c];
  else if (c < 333) v = lh[b * 135 + (c - 198)];
  else if (c < 468) v = rh[b * 135 + (c - 333)];
  else if (c < FEAT) v = shp[b * 11 + (c - 468)];
  else v = 0.0f;                       // zero pad column 479
  feat[idx] = v;
}

// Replicates reference _edit_pose on batch 0 (note the reference's
// roll/yaw swap: e0(roll)->"yaw", e2(yaw)->"roll"), pitch override 20deg
// only on body joint 0 (j==0).
__global__ void edit_pose0_kernel(float* __restrict__ feat) {
  int j = threadIdx.x;
  if (j >= 52) return;
  float* f = feat + j * 9;   // batch row 0
  float R00 = f[0];
  float R10 = f[3];
  float R20 = f[6], R21 = f[7], R22 = f[8];
  float roll  = atan2f(R21, R22);
  float syv   = sqrtf(R00 * R00 + R10 * R10);
  float pitch = (syv < 1e-6f) ? atan2f(-R20, 1e-6f) : atan2f(-R20, syv);
  float yaw   = atan2f(R10, R00);
  // reference swap: Y := e[...,0] (roll), P := e[...,1], R := e[...,2] (yaw)
  float Y = roll;
  float P = (j == 0) ? (20.0f * 3.14159265358979323846f / 180.0f) : pitch;
  float Rr = yaw;
  float cy = cosf(Y),  sy = sinf(Y);
  float cp = cosf(P),  sp = sinf(P);
  float cr = cosf(Rr), sr = sinf(Rr);
  // Rz(Y) @ Ry(P) @ Rx(Rr)
  f[0] = cy * cp;  f[1] = -sy * cr + cy * sp * sr;  f[2] =  sy * sr + cy * sp * cr;
  f[3] = sy * cp;  f[4] =  cy * cr + sy * sp * sr;  f[5] = -cy * sr + sy * sp * cr;
  f[6] = -sp;      f[7] =  cp * sr;                 f[8] =  cp * cr;
}

// ---------------- WMMA GEMM: out[1024 x N] = feat[1024 x 480] @ W[479 x N] ----
// block = 256 threads (8 wave32), grid = (ceil(N/128), 64)
// each wave computes one 16x16 fp32 tile with V_WMMA_F32_16X16X4_F32.
// N is a compile-time constant so B-fragment loads / D stores use immediate
// offsets instead of per-iteration 64-bit multiplies.
template <int N>
__global__ __launch_bounds__(256)
void gemm_wmma_kernel(const float* __restrict__ feat,
                      const float* __restrict__ W,
                      float* __restrict__ out) {
  __shared__ float lf[16 * KPAD];        // 30 KB feat M-tile

  const int tid  = threadIdx.x;
  const int lane = tid & 31;
  const int wave = tid >> 5;             // 0..7
  const int m0   = blockIdx.y * 16;
  const int n0   = blockIdx.x * 128 + wave * 16;

  // stage 16 rows of feat into LDS (contiguous, 128-bit chunks)
  {
    const float4* fsrc = (const float4*)(feat + (size_t)m0 * KPAD);
    float4* ldst = (float4*)lf;
    for (int idx = tid; idx < (16 * KPAD) / 4; idx += 256) ldst[idx] = fsrc[idx];
  }
  __syncthreads();

  // fragment coordinates (wave32 layouts, CDNA5 ISA §7.12.2)
  const int mrow  = lane & 15;           // A: M = lane%16
  const int khalf = (lane >> 4) * 2;     // A/B: K offset 0 or 2 per half-wave
  const int n     = n0 + (lane & 15);    // B/D: N = lane%16
  const bool nvalid = (n < N);
  const int nclamp  = nvalid ? n : (N - 1);

  // A fragments: two consecutive K values per lane from LDS
  const v2f* lfa = (const v2f*)(lf + mrow * KPAD + khalf);
  // B fragments: rows khalf, khalf+1 at column nclamp; advance 4 rows/step
  const float* p = W + (size_t)khalf * N + nclamp;

  v8f acc0 = {};
  v8f acc1 = {};
  // main loop: K-groups 0..118 (rows 0..475), no clamping needed
#pragma unroll 4
  for (int k0 = 0; k0 < KPAD - 4; k0 += 4) {
    v2f a = lfa[k0 >> 1];
    if ((k0 & 15) == 0 && (k0 + 34) < FEAT) {
      __builtin_prefetch(p + (size_t)32 * N, 0, 1);   // global_prefetch_b8
    }
    v2f b;
    b.x = p[0];
    b.y = p[N];
    p += (size_t)4 * N;
    if ((k0 & 4) == 0) {
      acc0 = __builtin_amdgcn_wmma_f32_16x16x4_f32(false, a, false, b,
                                                   (short)0, acc0, false, false);
    } else {
      acc1 = __builtin_amdgcn_wmma_f32_16x16x4_f32(false, a, false, b,
                                                   (short)0, acc1, false, false);
    }
  }
  // tail K-group (rows 476..479; row 479 is the zero-pad column):
  // lanes with khalf==2 must not touch row 479 -> clamp address, zero value.
  {
    v2f a = lfa[(KPAD - 4) >> 1];        // LDS element 479 is 0 (padded)
    float by = p[(khalf == 0) ? N : 0];  // row 477 (valid) or safe re-read
    v2f b;
    b.x = p[0];                          // rows 476 / 478, both valid
    b.y = (khalf == 0) ? by : 0.0f;
    acc0 = __builtin_amdgcn_wmma_f32_16x16x4_f32(false, a, false, b,
                                                 (short)0, acc0, false, false);
  }
  v8f acc = acc0 + acc1;

  if (nvalid) {
    const int mhalf = (lane >> 4) * 8;   // D: M = vgpr + 8*(lane/16)
    float* op = out + (size_t)(m0 + mhalf) * N + n;
#pragma unroll
    for (int v = 0; v < 8; ++v) {
      // write-once streaming output: keep it out of L2 so W stays resident
      __builtin_nontemporal_store(acc[v], op + (size_t)v * N);
    }
  }
}

// ---------------- camera translation ----------------
__global__ void camt_kernel(const float* __restrict__ cam,
                            const float* __restrict__ Kin,
                            const float* __restrict__ bscale,
                            const float* __restrict__ bcenter,
                            const int* __restrict__ img_w,
                            const int* __restrict__ img_h,
                            float* __restrict__ ct_ws,
                            float* __restrict__ out_ct) {
  int b = blockIdx.x * blockDim.x + threadIdx.x;
  if (b >= BATCH) return;
  float s  = cam[b * 3 + 0];
  float tx = cam[b * 3 + 1];
  float ty = cam[b * 3 + 2];
  float focal = Kin[b * 9 + 0];
  float bh = bscale[b] * 200.0f;
  // tz = 2*focal/(r*224*s), r = bh/224  ->  2*focal/(bh*s)
  float tz = 2.0f * focal / (bh * s);
  float w = (float)img_w[0];
  float h = (float)img_h[0];
  float cx = 2.0f * (bcenter[b * 2 + 0] - w * 0.5f) / (s * bh);
  float cy = 2.0f * (bcenter[b * 2 + 1] - h * 0.5f) / (s * bh);
  float t0 = tx + cx, t1 = ty + cy, t2 = tz;
  ct_ws[b * 3 + 0] = t0;  ct_ws[b * 3 + 1] = t1;  ct_ws[b * 3 + 2] = t2;
  out_ct[b * 3 + 0] = t0; out_ct[b * 3 + 1] = t1; out_ct[b * 3 + 2] = t2;
}

// ---------------- joints: +0.5 z, write joints3d, project joints2d --------
__global__ void project_kernel(const float* __restrict__ jraw,
                               const float* __restrict__ ct,
                               const float* __restrict__ Kin,
                               float* __restrict__ out_j3,
                               float* __restrict__ out_j2) {
  int idx = blockIdx.x * blockDim.x + threadIdx.x;
  if (idx >= BATCH * NJ) return;
  int b = idx / NJ;
  int j = idx - b * NJ;
  const float* jr = jraw + (size_t)b * NJCOL + j * 3;
  float x = jr[0], y = jr[1], z = jr[2] + 0.5f;
  float* j3 = out_j3 + (size_t)idx * 3;
  j3[0] = x; j3[1] = y; j3[2] = z;
  float px = x + ct[b * 3 + 0];
  float py = y + ct[b * 3 + 1];
  float pz = z + ct[b * 3 + 2];
  float inv = 1.0f / pz;
  float nx = px * inv, ny = py * inv;   // normalized z == 1
  const float* K = Kin + b * 9;
  out_j2[(size_t)idx * 2 + 0] = K[0] * nx + K[1] * ny + K[2];
  out_j2[(size_t)idx * 2 + 1] = K[3] * nx + K[4] * ny + K[5];
}

// ---------------- launch ----------------
extern "C" void kernel_launch(void* const* d_in, const int* in_sizes, int n_in,
                              void* d_out, int out_size, void* d_ws, size_t ws_size,
                              hipStream_t stream) {
  const float* body    = (const float*)d_in[0];   // (B,22,3,3)
  const float* lh      = (const float*)d_in[1];   // (B,15,3,3)
  const float* rh      = (const float*)d_in[2];   // (B,15,3,3)
  const float* shp     = (const float*)d_in[3];   // (B,11)
  const float* cam     = (const float*)d_in[4];   // (B,3)
  const float* Kin     = (const float*)d_in[5];   // (B,3,3)
  const float* bscale  = (const float*)d_in[6];   // (B,)
  const float* bcenter = (const float*)d_in[7];   // (B,2)
  const int*   img_w   = (const int*)d_in[8];     // scalar
  const int*   img_h   = (const int*)d_in[9];     // scalar
  const float* W_j     = (const float*)d_in[10];  // (479, 381)
  const float* W_v     = (const float*)d_in[11];  // (479, 31425)

  float* out = (float*)d_out;
  // output layout: vertices | joints3d | joints2d | cam_t
  float* out_v  = out;
  float* out_j3 = out + (size_t)BATCH * NVCOL;
  float* out_j2 = out_j3 + (size_t)BATCH * NJCOL;
  float* out_ct = out_j2 + (size_t)BATCH * NJ * 2;

  // workspace layout (floats): feat[B*480] | jraw[B*381] | cam_t[B*3]
  float* feat = (float*)d_ws;
  float* jraw = feat + (size_t)BATCH * KPAD;
  float* ctws = jraw + (size_t)BATCH * NJCOL;

  // 1) assemble feat (zero-padded K)
  {
    int total = BATCH * KPAD;
    build_feat_kernel<<<(total + 255) / 256, 256, 0, stream>>>(body, lh, rh, shp, feat);
  }
  // 2) euler-edit the 52 joints of batch 0 in place
  edit_pose0_kernel<<<1, 64, 0, stream>>>(feat);
  // 3) camera translation
  camt_kernel<<<(BATCH + 255) / 256, 256, 0, stream>>>(cam, Kin, bscale, bcenter,
                                                       img_w, img_h, ctws, out_ct);
  // 4) big GEMM: vertices (N = 31425)
  {
    dim3 grid((NVCOL + 127) / 128, BATCH / 16);
    gemm_wmma_kernel<NVCOL><<<grid, 256, 0, stream>>>(feat, W_v, out_v);
  }
  // 5) small GEMM: joints3d raw (N = 381)
  {
    dim3 grid((NJCOL + 127) / 128, BATCH / 16);
    gemm_wmma_kernel<NJCOL><<<grid, 256, 0, stream>>>(feat, W_j, jraw);
  }
  // 6) joints3d (+0.5 z) and perspective projection
  {
    int total = BATCH * NJ;
    project_kernel<<<(total + 255) / 256, 256, 0, stream>>>(jraw, ctws, Kin,
                                                            out_j3, out_j2);
  }
}